// ArcFaceLoss_43473658970232
// MI455X (gfx1250) — compile-verified
//
#include <hip/hip_runtime.h>
#include <stdint.h>

// ---------------- problem constants (reference: N=8192, C=32000) ----------
#define CDIM      32000
#define NF4       (CDIM / 4)            // 8000 float4 per row
#define TILE_F4   512                   // float4 per LDS tile (8 KB)
#define FULL_TILES 15                   // 15 * 512 = 7680 float4 via async LDS
#define TAIL_BASE (FULL_TILES * TILE_F4)// remaining 320 float4 via direct loads
#define TPB       256                   // 8 waves (wave32); 512/256 == 2 exactly

static_assert(TILE_F4 == 2 * TPB, "each thread copies exactly 2 float4/tile");
static_assert(TAIL_BASE <= NF4, "tail must be non-negative");

#define S_SCALE 30.0f
#define COS_M   0.8775825618903728f     // cos(0.5)
#define SIN_M   0.4794255386042030f     // sin(0.5)
#define MM_C    0.2397127693021015f     // sin(pi - 0.5) * 0.5
#define THRESH  (-0.8775825618903728f)  // cos(pi - 0.5)

// -------- gfx1250 async global->LDS path (guarded; fallback compiles clean)
#if defined(__gfx1250__) && \
    __has_builtin(__builtin_amdgcn_global_load_async_to_lds_b128) && \
    __has_builtin(__builtin_amdgcn_s_wait_asynccnt)
#define USE_ASYNC 1
#else
#define USE_ASYNC 0
#endif

#if USE_ASYNC
typedef int v4i __attribute__((ext_vector_type(4)));
typedef __attribute__((address_space(3))) v4i lds_v4i;

// Copy 16 bytes global -> LDS via the async DMA path (tracked by ASYNCcnt).
// Pointer args go through integer casts so only the *type* changes: generic
// pointers to global memory are value-identical to AS1/generic, and generic
// LDS pointers carry the workgroup-relative LDS byte offset in their low 32
// bits (flat->LDS mapping truncates to addr[31:0] per the CDNA5 aperture
// rules).
__device__ __forceinline__ void async_cp16(const void* g, const void* l) {
    __builtin_amdgcn_global_load_async_to_lds_b128(
        (v4i*)(uintptr_t)g,
        (lds_v4i*)(uint32_t)(uintptr_t)l,
        0, 0);
}
#endif

__device__ __forceinline__ float clamp1(float v) {
#if __has_builtin(__builtin_amdgcn_fmed3f)
    return __builtin_amdgcn_fmed3f(v, -1.0f, 1.0f);   // single v_med3_f32
#else
    return fminf(1.0f, fmaxf(-1.0f, v));
#endif
}

__device__ __forceinline__ float esum4(float4 a) {
    // logits bounded in [-30,30] -> exp never overflows fp32; no max pass.
    return __expf(S_SCALE * clamp1(a.x)) + __expf(S_SCALE * clamp1(a.y))
         + __expf(S_SCALE * clamp1(a.z)) + __expf(S_SCALE * clamp1(a.w));
}

// One workgroup per row: stream 128 KB through LDS (double buffered async DMA;
// every wave issues exactly 2 exec-uniform async b128 ops per tile, so
// s_wait_asynccnt(2) retires precisely the current tile while the next is in
// flight). The 320-float4 tail is read directly into registers. Accumulate
// sum of exp(S*clamp(x)), patch the target column, emit per-row NLL into d_ws.
__global__ void __launch_bounds__(TPB)
arcface_row_kernel(const float* __restrict__ pred,
                   const int*   __restrict__ target,
                   float*       __restrict__ row_loss)
{
    const int row = blockIdx.x;
    const int tid = threadIdx.x;
    const float*  rowp = pred + (size_t)row * CDIM;
    const float4* row4 = (const float4*)rowp;

    float acc = 0.0f;

#if USE_ASYNC
    __shared__ float4 buf[2][TILE_F4];   // 2 x 8 KB double buffer

    // Prologue: kick off tile 0 (2 unconditional async issues per wave).
    async_cp16(row4 + tid,       &buf[0][tid]);
    async_cp16(row4 + tid + TPB, &buf[0][tid + TPB]);

    for (int t = 0; t < FULL_TILES; ++t) {
        if (t + 1 < FULL_TILES) {
            const float4* g = row4 + (t + 1) * TILE_F4;
            float4* lb = buf[(t + 1) & 1];
            async_cp16(g + tid,       &lb[tid]);
            async_cp16(g + tid + TPB, &lb[tid + TPB]);
            __builtin_amdgcn_s_wait_asynccnt(2);   // tile t landed; t+1 in flight
        } else {
            __builtin_amdgcn_s_wait_asynccnt(0);   // drain everything
        }
        __syncthreads();                            // tile t visible to all waves

        const float4* lb = buf[t & 1];
        acc += esum4(lb[tid]);
        acc += esum4(lb[tid + TPB]);

        __syncthreads();                            // buffer safe to overwrite
    }

    // Tail: 320 float4 straight from global (no LDS, no counter interaction).
    for (int i = TAIL_BASE + tid; i < NF4; i += TPB)
        acc += esum4(row4[i]);
#else
    for (int i = tid; i < NF4; i += TPB)
        acc += esum4(row4[i]);
#endif

    // Deterministic block tree reduction.
    __shared__ float red[TPB];
    red[tid] = acc;
    __syncthreads();
    for (int s = TPB / 2; s > 0; s >>= 1) {
        if (tid < s) red[tid] += red[tid + s];
        __syncthreads();
    }

    if (tid == 0) {
        const int t = target[row];
        const float tc = clamp1(rowp[t]);
        float tm;
        if (tc > THRESH) {
            // cos(acos(x) + M) = x*cosM - sqrt(1-x^2)*sinM
            tm = tc * COS_M - sqrtf(fmaxf(0.0f, 1.0f - tc * tc)) * SIN_M;
        } else {
            tm = tc - MM_C;
        }
        // swap the plain target term for the margined one, then NLL:
        // loss = log(sum_j exp(S*logit_j)) - S*logit_target
        const float sum = red[0] - __expf(S_SCALE * tc) + __expf(S_SCALE * tm);
        row_loss[row] = __logf(sum) - S_SCALE * tm;
    }
}

__global__ void __launch_bounds__(TPB)
arcface_finalize_kernel(const float* __restrict__ row_loss,
                        float*       __restrict__ out,
                        int n_rows)
{
    __shared__ float red[TPB];
    float s = 0.0f;
    for (int i = threadIdx.x; i < n_rows; i += TPB)
        s += row_loss[i];
    red[threadIdx.x] = s;
    __syncthreads();
    for (int k = TPB / 2; k > 0; k >>= 1) {
        if (threadIdx.x < k) red[threadIdx.x] += red[threadIdx.x + k];
        __syncthreads();
    }
    if (threadIdx.x == 0)
        out[0] = red[0] / (float)n_rows;   // LOSS_WEIGHT == 1.0
}

extern "C" void kernel_launch(void* const* d_in, const int* in_sizes, int n_in,
                              void* d_out, int out_size, void* d_ws, size_t ws_size,
                              hipStream_t stream)
{
    const float* pred   = (const float*)d_in[0];
    const int*   target = (const int*)d_in[1];
    float*       out    = (float*)d_out;

    const int n_rows = in_sizes[1];          // 8192; C is compile-time 32000
    float* row_loss = (float*)d_ws;          // 8192 * 4 B scratch, fully written

    arcface_row_kernel<<<n_rows, TPB, 0, stream>>>(pred, target, row_loss);
    arcface_finalize_kernel<<<1, TPB, 0, stream>>>(row_loss, out, n_rows);
}